// AttentionST_3513283248325
// MI455X (gfx1250) — compile-verified
//
#include <hip/hip_runtime.h>
#include <hip/hip_bf16.h>

typedef __attribute__((ext_vector_type(2))) float v2f;
typedef __attribute__((ext_vector_type(8))) float v8f;

#define BATCH  256
#define SEQ    512
#define DIM    768
#define CHUNK  16                    // tokens staged per LDS chunk
#define NCHUNK (SEQ / CHUNK)         // 32
#define LROW   (DIM + 8)             // 776 floats: 2*LROW mod 64 == 16 -> conflict-free B tiles
#define CHUNK_FLOATS (CHUNK * LROW)  // 12416 floats per buffer

// Use raw CDNA5 asm for the async global->LDS path: the builtin's signature
// diverges between toolchains, the ISA encoding does not.
#if defined(__gfx1250__)
#define HAVE_ASYNC 1
#else
#define HAVE_ASYNC 0
#endif

#if HAVE_ASYNC
  #if __has_builtin(__builtin_amdgcn_s_wait_asynccnt)
    #define WAIT_ASYNC(n) __builtin_amdgcn_s_wait_asynccnt(n)
  #else
    #define WAIT_ASYNC(n) asm volatile("s_wait_asynccnt %0" :: "i"(n) : "memory")
  #endif
#else
  #define WAIT_ASYNC(n) ((void)0)
#endif

// Async-copy 16 bytes from global to LDS (per lane).
__device__ __forceinline__ void async_copy_b128(const void* gsrc, void* ldst) {
#if HAVE_ASYNC
    // VDST = LDS byte offset (low 32 bits of the flat address == AS3 offset),
    // VADDR = 64-bit global address, SADDR = off.
    asm volatile("global_load_async_to_lds_b128 %0, %1, off"
                 :: "v"((unsigned)(uintptr_t)ldst), "v"(gsrc)
                 : "memory");
#else
    *(float4*)ldst = *(const float4*)gsrc;
#endif
}

// Stage one 768-float token row into LDS: 16 threads per row, 12 float4 each.
__device__ __forceinline__ void stage_row(const float4* __restrict__ g4row,
                                          float* lrow, int sub) {
    float4* l4 = (float4*)lrow;
#pragma unroll
    for (int j = 0; j < 12; ++j)
        async_copy_b128(g4row + sub + 16 * j, l4 + sub + 16 * j);
}

// One workgroup per batch row; single pass over embeddings (flash-style online softmax).
__global__ __launch_bounds__(256) void attn_pool_flash(
    const float* __restrict__ emb,     // [B, S, D]
    const float* __restrict__ w_att,   // [D]
    const float* __restrict__ w_pred,  // [D]
    const float* __restrict__ b_pred,  // [1]
    float* __restrict__ out)           // [B]
{
    extern __shared__ __align__(16) float dynls[];   // 2 * CHUNK_FLOATS floats
    __shared__ float chunk_sc[CHUNK];
    __shared__ float chunk_w[CHUNK];
    __shared__ float redl[8];

    float* buf0 = dynls;
    float* buf1 = dynls + CHUNK_FLOATS;

    const int b    = blockIdx.x;
    const int tid  = threadIdx.x;
    const int w    = tid >> 5;          // wave 0..7
    const int lane = tid & 31;
    const int r    = tid >> 4;          // staging: token row 0..15
    const int sub  = tid & 15;          // staging: sub-slot in row
    const int koff = (lane >> 4) << 1;  // WMMA K offset: 0 lower half-wave, 2 upper
    const int nsub = lane & 15;         // WMMA N column

    const float4* E4 = (const float4*)(emb + (size_t)b * (SEQ * DIM));
    const float4* W4 = (const float4*)w_att;

    float4 wv[6];
#pragma unroll
    for (int j = 0; j < 6; ++j) wv[j] = W4[lane + 32 * j];

    v8f acc[6];
#pragma unroll
    for (int i = 0; i < 6; ++i) acc[i] = (v8f){0.f,0.f,0.f,0.f,0.f,0.f,0.f,0.f};
    float m_run = -3.0e38f;
    float L_run = 0.f;

    // Prestage chunk 0 into buf0.
    stage_row(E4 + r * (DIM / 4), buf0 + r * LROW, sub);

    for (int ci = 0; ci < NCHUNK; ++ci) {
        float* curbuf = (ci & 1) ? buf1 : buf0;
        float* nxtbuf = (ci & 1) ? buf0 : buf1;

        if (ci + 1 < NCHUNK) {
            // Overlap DMA of chunk ci+1 with compute on chunk ci.
            stage_row(E4 + (ci + 1) * CHUNK * (DIM / 4) + r * (DIM / 4),
                      nxtbuf + r * LROW, sub);
            WAIT_ASYNC(12);   // first 12 (chunk ci) done; next 12 may stay in flight
        } else {
            WAIT_ASYNC(0);
        }
        __syncthreads();      // chunk ci visible to all waves

        // ---- scores for the 16 staged tokens: wave w -> tokens 2w, 2w+1 ----
#pragma unroll
        for (int q = 0; q < 2; ++q) {
            const int tloc = w * 2 + q;
            const float4* rowp = (const float4*)(curbuf + tloc * LROW);
            float sum = 0.f;
#pragma unroll
            for (int j = 0; j < 6; ++j) {
                float4 e = rowp[lane + 32 * j];
                sum += e.x*wv[j].x + e.y*wv[j].y + e.z*wv[j].z + e.w*wv[j].w;
            }
#pragma unroll
            for (int off = 16; off >= 1; off >>= 1)
                sum += __shfl_xor(sum, off, 32);
            if (lane == 0) chunk_sc[tloc] = sum;
        }
        __syncthreads();

        // ---- online softmax update (replicated identically on every thread) ----
        float cmax = chunk_sc[0];
#pragma unroll
        for (int i = 1; i < CHUNK; ++i) cmax = fmaxf(cmax, chunk_sc[i]);
        const float m_new = fmaxf(m_run, cmax);
        const float alpha = __expf(m_run - m_new);   // exp(-huge) -> 0 on first chunk
        float csum = 0.f;
#pragma unroll
        for (int i = 0; i < CHUNK; ++i) csum += __expf(chunk_sc[i] - m_new);
        L_run = L_run * alpha + csum;
        m_run = m_new;
        if (tid < CHUNK) chunk_w[tid] = __expf(chunk_sc[tid] - m_new);
        __syncthreads();

        // ---- rescale accumulators + WMMA accumulate this chunk from LDS ----
        // V_WMMA_F32_16X16X4_F32: A[m][k] = chunk_w[k] (replicated rows),
        // B[k][n] = curbuf[k][d0+n]; result C[m][n] = partial pooled[d0+n].
#pragma unroll
        for (int i = 0; i < 6; ++i) {
            v8f c = acc[i] * alpha;
            const int d0 = (w + 8 * i) * 16;
#pragma unroll
            for (int k0 = 0; k0 < CHUNK; k0 += 4) {
                v2f a, bb;
                a[0] = chunk_w[k0 + koff];
                a[1] = chunk_w[k0 + koff + 1];
                const float* p = curbuf + (k0 + koff) * LROW + d0 + nsub;
                bb[0] = p[0];        // row k0+koff
                bb[1] = p[LROW];     // row k0+koff+1
#if __has_builtin(__builtin_amdgcn_wmma_f32_16x16x4_f32)
                c = __builtin_amdgcn_wmma_f32_16x16x4_f32(
                        false, a, false, bb, (short)0, c, false, false);
#else
                c[0] += a[0] * bb[0] + a[1] * bb[1];
#endif
            }
            acc[i] = c;
        }
        __syncthreads();   // all reads of curbuf done before it gets re-staged
    }

    // ---- fold pooled/L into logit, reduce, sigmoid ----
    const float invL = 1.0f / L_run;
    float logit_part = 0.f;
#pragma unroll
    for (int i = 0; i < 6; ++i) {
        float pooled;
#if __has_builtin(__builtin_amdgcn_wmma_f32_16x16x4_f32)
        pooled = acc[i][0];                       // pooled[d0 + (lane&15)] (rows replicated)
#else
        pooled = acc[i][0] + __shfl_xor(acc[i][0], 16, 32);
#endif
        if (lane < 16) logit_part += pooled * invL * w_pred[(w + 8 * i) * 16 + lane];
    }
#pragma unroll
    for (int off = 16; off >= 1; off >>= 1)
        logit_part += __shfl_xor(logit_part, off, 32);
    if (lane == 0) redl[w] = logit_part;
    __syncthreads();
    if (tid == 0) {
        float lg = b_pred[0];
#pragma unroll
        for (int i = 0; i < 8; ++i) lg += redl[i];
        out[b] = 1.0f / (1.0f + expf(-lg));
    }
}

extern "C" void kernel_launch(void* const* d_in, const int* in_sizes, int n_in,
                              void* d_out, int out_size, void* d_ws, size_t ws_size,
                              hipStream_t stream) {
    (void)in_sizes; (void)n_in; (void)out_size; (void)d_ws; (void)ws_size;
    const float* emb    = (const float*)d_in[0];
    const float* w_att  = (const float*)d_in[1];
    const float* w_pred = (const float*)d_in[2];
    const float* b_pred = (const float*)d_in[3];
    float* out = (float*)d_out;

    const size_t dyn_lds = 2 * CHUNK_FLOATS * sizeof(float);   // ~97 KB
    dim3 grid(BATCH), block(256);
    attn_pool_flash<<<grid, block, dyn_lds, stream>>>(emb, w_att, w_pred, b_pred, out);
}